// Net_58334245814278
// MI455X (gfx1250) — compile-verified
//
#include <hip/hip_runtime.h>

#define Bn  64
#define Nn  116
#define Np  128
#define NT  8      // 128/16 tiles per side
#define Kk  10
#define DIL 2

typedef __attribute__((ext_vector_type(2))) float v2f;
typedef __attribute__((ext_vector_type(8))) float v8f;

// ---------------------------------------------------------------------------
// Generic batched 128x128x128 f32 GEMM on WMMA 16x16x4. One wave per 16x16
// C tile. TRANSB=true computes A * B^T (used for the Gram matrix X*X^T).
// strideB==0 broadcasts a single B across the batch (weight projections).
// ---------------------------------------------------------------------------
template <bool TRANSB>
__global__ void gemm128_wmma(const float* __restrict__ A, long sA,
                             const float* __restrict__ Bm, long sB,
                             float* __restrict__ C, long sC, int nbatch) {
  int wave = (int)((blockIdx.x * blockDim.x + threadIdx.x) >> 5);
  int lane = threadIdx.x & 31;
  int ntiles = nbatch * NT * NT;
  if (wave >= ntiles) return;               // whole-wave uniform -> EXEC all 1s
  int b  = wave / (NT * NT);
  int t  = wave % (NT * NT);
  int ti = (t / NT) * 16;
  int tj = (t % NT) * 16;
  const float* Ab = A  + (long)b * sA;
  const float* Bb = Bm + (long)b * sB;
  float*       Cb = C  + (long)b * sC;
  int hi = lane >> 4;                       // 0 or 1
  int lo = lane & 15;
  v8f acc = {};
#pragma unroll 4
  for (int k0 = 0; k0 < Np; k0 += 4) {
    v2f a, bv;
    // A 16x4 tile: lane(0-15)=row M, VGPR0/1 = K{0,1} (lo half) or K{2,3} (hi half)
    a.x = Ab[(ti + lo) * Np + k0 + 2 * hi];
    a.y = Ab[(ti + lo) * Np + k0 + 2 * hi + 1];
    if (TRANSB) {
      bv.x = Bb[(tj + lo) * Np + k0 + hi];       // B_eff[k][n] = Bsrc[n][k]
      bv.y = Bb[(tj + lo) * Np + k0 + 2 + hi];
    } else {
      bv.x = Bb[(k0 + hi) * Np + tj + lo];       // B 4x16: row K striped on lanes
      bv.y = Bb[(k0 + 2 + hi) * Np + tj + lo];
    }
    acc = __builtin_amdgcn_wmma_f32_16x16x4_f32(
        /*neg_a=*/false, a, /*neg_b=*/false, bv,
        /*c_mod=*/(short)0, acc, /*reuse_a=*/false, /*reuse_b=*/false);
  }
#pragma unroll
  for (int r = 0; r < 8; ++r)
    Cb[(ti + r + 8 * hi) * Np + tj + lo] = acc[r];
}

// ---------------------------------------------------------------------------
// Padding / elementwise helpers
// ---------------------------------------------------------------------------
__global__ void pad_x(const float* __restrict__ x, float* __restrict__ Xp) {
  int idx = blockIdx.x * blockDim.x + threadIdx.x;
  if (idx >= Bn * Np * Np) return;
  int b = idx / (Np * Np), r = (idx / Np) % Np, c = idx % Np;
  float v = 0.f;
  if (r < Nn && c < Nn) v = x[((long)b * Nn + r) * Nn + c];
  Xp[idx] = v;
}

// Split conv weight (2N x Nn) into padded Wa, Wb (128x128 each)
__global__ void pad_w(const float* __restrict__ W, float* __restrict__ Wa,
                      float* __restrict__ Wb) {
  int idx = blockIdx.x * blockDim.x + threadIdx.x;
  if (idx >= Np * Np) return;
  int r = idx / Np, c = idx % Np;
  float va = 0.f, vb = 0.f;
  if (r < Nn && c < Nn) {
    va = W[r * Nn + c];
    vb = W[(Nn + r) * Nn + c];
  }
  Wa[idx] = va; Wb[idx] = vb;
}

// d2[i][j] = G[i][i] + G[j][j] - 2 G[i][j]  (+1e10 on diagonal)
__global__ void make_d2(const float* __restrict__ G, float* __restrict__ d2) {
  int idx = blockIdx.x * blockDim.x + threadIdx.x;
  if (idx >= Bn * Np * Np) return;
  int b = idx / (Np * Np), i = (idx / Np) % Np, j = idx % Np;
  float v = 1e30f;
  if (i < Nn && j < Nn) {
    const float* Gb = G + (long)b * Np * Np;
    v = Gb[i * Np + i] + Gb[j * Np + j] - 2.f * Gb[i * Np + j];
    if (i == j) v += 1e10f;
  }
  d2[idx] = v;
}

// 20 smallest per row (selection), keep sorted positions 0,2,...,18 (dilation 2)
__global__ void knn_topk(const float* __restrict__ d2, float* __restrict__ M1) {
  int row = blockIdx.x * blockDim.x + threadIdx.x;
  if (row >= Bn * Nn) return;
  int b = row / Nn, i = row % Nn;
  const float* r = d2 + ((long)b * Np + i) * Np;
  float loc[Nn];
  for (int j = 0; j < Nn; ++j) loc[j] = r[j];
  float* Mrow = M1 + ((long)b * Np + i) * Np;
  for (int t = 0; t < Kk * DIL; ++t) {
    int arg = 0; float best = loc[0];
    for (int j = 1; j < Nn; ++j)
      if (loc[j] < best) { best = loc[j]; arg = j; }
    loc[arg] = 1e30f;
    if ((t & 1) == 0) Mrow[arg] = 1.f;
  }
}

// two_hop: out[i][j] = (P[j][i] > 0 && M[j][i] == 0), P = M @ M
__global__ void twohop_fin(const float* __restrict__ P,
                           const float* __restrict__ M, float* __restrict__ out) {
  int idx = blockIdx.x * blockDim.x + threadIdx.x;
  if (idx >= Bn * Np * Np) return;
  int b = idx / (Np * Np), i = (idx / Np) % Np, j = idx % Np;
  const float* Pb = P + (long)b * Np * Np;
  const float* Mb = M + (long)b * Np * Np;
  float p = Pb[j * Np + i];
  float m = Mb[j * Np + i];
  out[idx] = (p > 0.f && m < 0.5f) ? 1.f : 0.f;
}

__global__ void scatter_adj(const int* __restrict__ edge, int nedges,
                            float* __restrict__ Min) {
  int e = blockIdx.x * blockDim.x + threadIdx.x;
  if (e >= nedges) return;
  int src = edge[e];
  int tgt = edge[nedges + e];
  int bb = tgt / Nn, ti = tgt % Nn, sj = src % Nn;
  Min[((long)bb * Np + ti) * Np + sj] = 1.f;  // duplicate writes of 1.0 are fine
}

// ---------------------------------------------------------------------------
// edge_conv (F_out = 116): masked max over neighbors, fused batch-norm + acc
// block = (b,i), threads over feature f
// ---------------------------------------------------------------------------
__global__ void edge_conv_acc(const float* __restrict__ xa,
                              const float* __restrict__ xb,
                              const float* __restrict__ M,
                              const float* __restrict__ bias,
                              const float* __restrict__ g,
                              const float* __restrict__ bb2,
                              float* __restrict__ acc, int accumulate) {
  int b = blockIdx.x / Nn, i = blockIdx.x % Nn;
  int f = threadIdx.x;
  __shared__ float mrow[Nn];
  __shared__ float xbi[Np];
  const float* Mrow = M + ((long)b * Np + i) * Np;
  for (int j = threadIdx.x; j < Nn; j += blockDim.x) mrow[j] = Mrow[j];
  const float* xbb = xb + (long)b * Np * Np;
  if (f < Np) xbi[f] = xbb[i * Np + f];
  __syncthreads();
  if (f >= Nn) return;
  float pre = xa[((long)b * Np + i) * Np + f] - xbi[f] + bias[f];
  float mx = -3.402823e38f; int has = 0;
  for (int j = 0; j < Nn; ++j) {
    if (mrow[j] > 0.f) {
      float v = pre + xbb[j * Np + f];
      v = v > 0.f ? v : 0.f;
      mx = v > mx ? v : mx;
      has = 1;
    }
  }
  float out = has ? mx : 0.f;
  float res = out * (g[f] * rsqrtf(1.00001f)) + bb2[f];
  float* dst = acc + ((long)b * Np + i) * Np + f;
  if (accumulate) *dst += res; else *dst = res;
}

// F_out = 1 projection: xa2[b][i] = X[b,i,:]·W[:116], xb2 with W[116:]
__global__ void proj1(const float* __restrict__ Xp, const float* __restrict__ W,
                      float* __restrict__ xa2, float* __restrict__ xb2) {
  int row = blockIdx.x * blockDim.x + threadIdx.x;
  if (row >= Bn * Nn) return;
  int b = row / Nn, i = row % Nn;
  const float* xr = Xp + ((long)b * Np + i) * Np;
  float sa = 0.f, sb = 0.f;
  for (int k = 0; k < Nn; ++k) {
    float xv = xr[k];
    sa += xv * W[k];
    sb += xv * W[Nn + k];
  }
  xa2[row] = sa; xb2[row] = sb;
}

__global__ void edge_conv1_acc(const float* __restrict__ xa2,
                               const float* __restrict__ xb2,
                               const float* __restrict__ M,
                               const float* __restrict__ bias,
                               const float* __restrict__ g,
                               const float* __restrict__ bb2,
                               float* __restrict__ acc, int accumulate) {
  int row = blockIdx.x * blockDim.x + threadIdx.x;
  if (row >= Bn * Nn) return;
  int b = row / Nn, i = row % Nn;
  const float* Mrow = M + ((long)b * Np + i) * Np;
  const float* xbb = xb2 + (long)b * Nn;
  float pre = xa2[row] - xbb[i] + bias[0];
  float mx = -3.402823e38f; int has = 0;
  for (int j = 0; j < Nn; ++j) {
    if (Mrow[j] > 0.f) {
      float v = pre + xbb[j];
      v = v > 0.f ? v : 0.f;
      mx = v > mx ? v : mx;
      has = 1;
    }
  }
  float out = has ? mx : 0.f;
  float res = out * (g[0] * rsqrtf(1.00001f)) + bb2[0];
  acc[row] = accumulate ? acc[row] + res : res;
}

// unpad x1 into output region
__global__ void write_x1(const float* __restrict__ x1acc, float* __restrict__ o) {
  int idx = blockIdx.x * blockDim.x + threadIdx.x;
  if (idx >= Bn * Nn * Nn) return;
  int b = idx / (Nn * Nn), i = (idx / Nn) % Nn, j = idx % Nn;
  o[idx] = x1acc[((long)b * Np + i) * Np + j];
}

// fea[b] = concat(max_i x1, mean_i x1, x_2)  (116+116+116 = 348)
__global__ void make_fea(const float* __restrict__ x1acc,
                         const float* __restrict__ x2acc, float* __restrict__ fea) {
  int b = blockIdx.x, f = threadIdx.x;
  if (f >= Nn) return;
  const float* base = x1acc + (long)b * Np * Np + f;
  float mx = -3.402823e38f, sm = 0.f;
  for (int i = 0; i < Nn; ++i) {
    float v = base[i * Np];
    mx = v > mx ? v : mx;
    sm += v;
  }
  fea[b * 348 + f]          = mx;
  fea[b * 348 + Nn + f]     = sm / (float)Nn;
  fea[b * 348 + 2 * Nn + f] = x2acc[b * Nn + f];
}

// tiny FC head: one thread per batch element
__global__ void fc_head(const float* __restrict__ fea, const float* __restrict__ pcd,
                        const float* __restrict__ fc1W, const float* __restrict__ fc1b,
                        const float* __restrict__ g4,   const float* __restrict__ b4,
                        const float* __restrict__ fc2W, const float* __restrict__ fc2b,
                        const float* __restrict__ g5,   const float* __restrict__ b5,
                        const float* __restrict__ fc3W, const float* __restrict__ fc3b,
                        float* __restrict__ probs, float* __restrict__ logits) {
  int b = blockIdx.x * blockDim.x + threadIdx.x;
  if (b >= Bn) return;
  const float* fr = fea + b * 348;
  float h[31];
  for (int o = 0; o < 24; ++o) {
    float s = fc1b[o];
    for (int k = 0; k < 348; ++k) s += fr[k] * fc1W[k * 24 + o];
    s = s > 0.f ? s : 0.f;
    h[o] = s * (g4[o] * rsqrtf(1.00001f)) + b4[o];
  }
  float nrm = 0.f;
  for (int k = 0; k < 7; ++k) { float v = pcd[b * 7 + k]; nrm += v * v; }
  nrm = sqrtf(nrm);
  nrm = nrm > 1e-12f ? nrm : 1e-12f;
  for (int k = 0; k < 7; ++k) h[24 + k] = pcd[b * 7 + k] / nrm;
  float h2[8];
  for (int o = 0; o < 8; ++o) {
    float s = fc2b[o];
    for (int k = 0; k < 31; ++k) s += h[k] * fc2W[k * 8 + o];
    s = s > 0.f ? s : 0.f;
    h2[o] = s * (g5[o] * rsqrtf(1.00001f)) + b5[o];
  }
  float l0 = fc3b[0], l1 = fc3b[1];
  for (int k = 0; k < 8; ++k) {
    l0 += h2[k] * fc3W[k * 2];
    l1 += h2[k] * fc3W[k * 2 + 1];
  }
  logits[b * 2] = l0; logits[b * 2 + 1] = l1;
  float m = l0 > l1 ? l0 : l1;
  float e0 = __expf(l0 - m), e1 = __expf(l1 - m);
  probs[b * 2]     = e0 / (e0 + e1);
  probs[b * 2 + 1] = e1 / (e0 + e1);
}

// ---------------------------------------------------------------------------
extern "C" void kernel_launch(void* const* d_in, const int* in_sizes, int n_in,
                              void* d_out, int out_size, void* d_ws, size_t ws_size,
                              hipStream_t stream) {
  const float* x      = (const float*)d_in[0];
  const int*   edge   = (const int*)d_in[1];
  const float* pcd    = (const float*)d_in[4];
  // params in dict insertion order
  const float* conv_W[6]  = {(const float*)d_in[5],  (const float*)d_in[7],
                             (const float*)d_in[9],  (const float*)d_in[11],
                             (const float*)d_in[13], (const float*)d_in[15]};
  const float* conv_b[6]  = {(const float*)d_in[6],  (const float*)d_in[8],
                             (const float*)d_in[10], (const float*)d_in[12],
                             (const float*)d_in[14], (const float*)d_in[16]};
  const float* bn1_g = (const float*)d_in[17]; const float* bn1_b = (const float*)d_in[18];
  const float* bn2_g = (const float*)d_in[19]; const float* bn2_b = (const float*)d_in[20];
  const float* bn3_g = (const float*)d_in[21]; const float* bn3_b = (const float*)d_in[22];
  const float* bn4_g = (const float*)d_in[23]; const float* bn4_b = (const float*)d_in[24];
  const float* bn5_g = (const float*)d_in[25]; const float* bn5_b = (const float*)d_in[26];
  const float* fc1W  = (const float*)d_in[27]; const float* fc1b  = (const float*)d_in[28];
  const float* fc2W  = (const float*)d_in[29]; const float* fc2b  = (const float*)d_in[30];
  const float* fc3W  = (const float*)d_in[31]; const float* fc3b  = (const float*)d_in[32];

  float* ws = (float*)d_ws;
  size_t off = 0;
  auto alloc = [&](size_t n) { float* p = ws + off; off += n; return p; };
  const long   SP = (long)Np * Np;
  const size_t BP = (size_t)Bn * Np * Np;
  float* Xp    = alloc(BP);
  float* G     = alloc(BP);
  float* d2    = alloc(BP);
  float* M1    = alloc(BP);
  float* M1_2  = alloc(BP);
  float* Min   = alloc(BP);
  float* Min2  = alloc(BP);
  float* P     = alloc(BP);
  float* xa    = alloc(BP);
  float* xb    = alloc(BP);
  float* x1acc = alloc(BP);
  float* Wa    = alloc((size_t)Np * Np);
  float* Wb    = alloc((size_t)Np * Np);
  float* xa2   = alloc((size_t)Bn * Nn);
  float* xb2   = alloc((size_t)Bn * Nn);
  float* x2acc = alloc((size_t)Bn * Nn);

  float* out        = (float*)d_out;
  float* out_probs  = out;
  float* out_x1     = out + Bn * 2;
  float* out_fea    = out + Bn * 2 + Bn * Nn * Nn;
  float* out_logits = out + Bn * 2 + Bn * Nn * Nn + Bn * 348;

  const int ELT = (int)BP;
  dim3 gElt((ELT + 255) / 256), bElt(256);
  int gemmBlocks = (Bn * NT * NT + 3) / 4;      // 4 waves / block

  // 1) pad input
  hipLaunchKernelGGL(pad_x, gElt, bElt, 0, stream, x, Xp);

  // 2) Gram matrix G = X X^T  (WMMA, transB)
  hipLaunchKernelGGL((gemm128_wmma<true>), dim3(gemmBlocks), dim3(128), 0, stream,
                     Xp, SP, Xp, SP, G, SP, Bn);
  hipLaunchKernelGGL(make_d2, gElt, bElt, 0, stream, G, d2);

  // 3) knn mask
  hipMemsetAsync(M1, 0, BP * sizeof(float), stream);
  hipLaunchKernelGGL(knn_topk, dim3((Bn * Nn + 127) / 128), dim3(128), 0, stream, d2, M1);

  // 4) two-hop of M1 (WMMA P = M1 @ M1)
  hipLaunchKernelGGL((gemm128_wmma<false>), dim3(gemmBlocks), dim3(128), 0, stream,
                     M1, SP, M1, SP, P, SP, Bn);
  hipLaunchKernelGGL(twohop_fin, gElt, bElt, 0, stream, P, M1, M1_2);

  // 5) dense adjacency + its two-hop
  int nedges = in_sizes[1] / 2;
  hipMemsetAsync(Min, 0, BP * sizeof(float), stream);
  hipLaunchKernelGGL(scatter_adj, dim3((nedges + 255) / 256), dim3(256), 0, stream,
                     edge, nedges, Min);
  hipLaunchKernelGGL((gemm128_wmma<false>), dim3(gemmBlocks), dim3(128), 0, stream,
                     Min, SP, Min, SP, P, SP, Bn);
  hipLaunchKernelGGL(twohop_fin, gElt, bElt, 0, stream, P, Min, Min2);

  // 6) four F=116 edge_convs accumulated into x1acc
  //    (conv idx, mask, bn_g, bn_b): conv1/M1/bn1, conv1_1/M1_2/bn1, conv3/Min/bn3, conv3_3/Min2/bn3
  const float* masks[4] = {M1, M1_2, Min, Min2};
  const int    cidx[4]  = {0, 1, 4, 5};
  const float* bng[4]   = {bn1_g, bn1_g, bn3_g, bn3_g};
  const float* bnb[4]   = {bn1_b, bn1_b, bn3_b, bn3_b};
  for (int c = 0; c < 4; ++c) {
    hipLaunchKernelGGL(pad_w, dim3((Np * Np + 255) / 256), dim3(256), 0, stream,
                       conv_W[cidx[c]], Wa, Wb);
    hipLaunchKernelGGL((gemm128_wmma<false>), dim3(gemmBlocks), dim3(128), 0, stream,
                       Xp, SP, Wa, 0L, xa, SP, Bn);
    hipLaunchKernelGGL((gemm128_wmma<false>), dim3(gemmBlocks), dim3(128), 0, stream,
                       Xp, SP, Wb, 0L, xb, SP, Bn);
    hipLaunchKernelGGL(edge_conv_acc, dim3(Bn * Nn), dim3(128), 0, stream,
                       xa, xb, masks[c], conv_b[cidx[c]], bng[c], bnb[c],
                       x1acc, c == 0 ? 0 : 1);
  }

  // 7) F=1 convs: conv2/M1 + conv2_2/M1_2 into x2acc
  hipLaunchKernelGGL(proj1, dim3((Bn * Nn + 127) / 128), dim3(128), 0, stream,
                     Xp, conv_W[2], xa2, xb2);
  hipLaunchKernelGGL(edge_conv1_acc, dim3((Bn * Nn + 127) / 128), dim3(128), 0, stream,
                     xa2, xb2, M1, conv_b[2], bn2_g, bn2_b, x2acc, 0);
  hipLaunchKernelGGL(proj1, dim3((Bn * Nn + 127) / 128), dim3(128), 0, stream,
                     Xp, conv_W[3], xa2, xb2);
  hipLaunchKernelGGL(edge_conv1_acc, dim3((Bn * Nn + 127) / 128), dim3(128), 0, stream,
                     xa2, xb2, M1_2, conv_b[3], bn2_g, bn2_b, x2acc, 1);

  // 8) outputs: x1, fea, FC head (probs + logits)
  hipLaunchKernelGGL(write_x1, dim3((Bn * Nn * Nn + 255) / 256), dim3(256), 0, stream,
                     x1acc, out_x1);
  hipLaunchKernelGGL(make_fea, dim3(Bn), dim3(128), 0, stream, x1acc, x2acc, out_fea);
  hipLaunchKernelGGL(fc_head, dim3(1), dim3(64), 0, stream,
                     out_fea, pcd, fc1W, fc1b, bn4_g, bn4_b,
                     fc2W, fc2b, bn5_g, bn5_b, fc3W, fc3b,
                     out_probs, out_logits);
}